// Quantizer_16793322127964
// MI455X (gfx1250) — compile-verified
//
#include <hip/hip_runtime.h>
#include <hip/hip_bf16.h>
#include <stdint.h>

// Quantize/dequantize: out = grid[nearest(x * gmax/alpha)] * alpha/gmax
// x: [C,F] f32, alpha: [C] f32, grid: [16] f32 sorted *uniform* (int4 grid).
//
// Memory-bound streaming kernel: 537 MB traffic -> ~23us at 23.3 TB/s.
// gfx1250 async global->LDS (ASYNCcnt) double-buffer pipeline + NT stores.

#define BLOCK_THREADS 256
#define VECS_PER_THREAD 4
#define VEC_PER_BLOCK (BLOCK_THREADS * VECS_PER_THREAD)  // 1024 float4 = 4096 floats

typedef float v4f __attribute__((ext_vector_type(4)));
typedef int   v4i __attribute__((ext_vector_type(4)));

#define AS1 __attribute__((address_space(1)))
#define AS3 __attribute__((address_space(3)))

#if defined(__AMDGCN__) && __has_builtin(__builtin_amdgcn_global_load_async_to_lds_b128)
#define HAVE_ASYNC_LDS 1
#else
#define HAVE_ASYNC_LDS 0
#endif

#if __has_builtin(__builtin_amdgcn_s_wait_asynccnt)
#define WAIT_ASYNC(n) __builtin_amdgcn_s_wait_asynccnt(n)
#else
#define WAIT_ASYNC(n) asm volatile("s_wait_asynccnt " #n ::: "memory")
#endif

// Snap 4 elements to nearest codebook entry (uniform grid) and dequantize.
// t = (x*scale - g0)/step folded into one fma: t = fma(x, mulA, addB).
__device__ __forceinline__ v4f snap4(v4f v, float mulA, float addB,
                                     float inv_scale, const float* s_grid) {
  v4f r;
#pragma unroll
  for (int k = 0; k < 4; ++k) {
    float t  = __builtin_fmaf(v[k], mulA, addB);
    float fi = __builtin_rintf(t);                      // v_rndne_f32
    fi = fminf(fmaxf(fi, 0.0f), 15.0f);                 // clamp to [0,15]
    float q = s_grid[(int)fi];                          // ds_load_b32: exact codebook value
    r[k] = q * inv_scale;
  }
  return r;
}

__global__ __launch_bounds__(BLOCK_THREADS) void quant_dequant_kernel(
    const float* __restrict__ x, const float* __restrict__ alpha,
    const float* __restrict__ grid, float* __restrict__ out, int vec_per_row) {
  __shared__ float s_grid[16];
#if HAVE_ASYNC_LDS
  __shared__ v4f s_tile[2][BLOCK_THREADS];  // 8 KB double buffer
#endif

  const int tid = threadIdx.x;
  if (tid < 16) s_grid[tid] = grid[tid];
  __syncthreads();

  const int row = blockIdx.y;           // uniform -> alpha via scalar load
  const float a   = alpha[row];
  const float g0  = s_grid[0];
  const float g15 = s_grid[15];                         // == max(grid)
  const float step      = (g15 - g0) * (1.0f / 15.0f);  // uniform grid spacing
  const float inv_step  = 1.0f / step;
  const float scale     = g15 / a;                      // gmax / alpha
  const float inv_scale = a / g15;                      // 1/scale
  const float mulA = scale * inv_step;
  const float addB = -g0 * inv_step;

  const int vec0 = blockIdx.x * VEC_PER_BLOCK + tid;
  const size_t row_base = (size_t)row * (size_t)vec_per_row;
  const v4f* __restrict__ xin  = (const v4f*)x + row_base;
  v4f* __restrict__ xout       = (v4f*)out + row_base;

#if HAVE_ASYNC_LDS
  // ---- async global->LDS double-buffered pipeline (ASYNCcnt) ----
  if (vec0 < vec_per_row)
    __builtin_amdgcn_global_load_async_to_lds_b128(
        (AS1 v4i*)(xin + vec0), (AS3 v4i*)&s_tile[0][tid], 0, 0);
#pragma unroll
  for (int i = 0; i < VECS_PER_THREAD; ++i) {
    if (i + 1 < VECS_PER_THREAD) {
      if (vec0 + (i + 1) * BLOCK_THREADS < vec_per_row)
        __builtin_amdgcn_global_load_async_to_lds_b128(
            (AS1 v4i*)(xin + vec0 + (i + 1) * BLOCK_THREADS),
            (AS3 v4i*)&s_tile[(i + 1) & 1][tid], 0, 0);
      WAIT_ASYNC(1);  // async loads complete in order: stage i is done
    } else {
      WAIT_ASYNC(0);
    }
    asm volatile("" ::: "memory");  // keep LDS read after the wait
    if (vec0 + i * BLOCK_THREADS < vec_per_row) {
      v4f v = s_tile[i & 1][tid];
      v4f r = snap4(v, mulA, addB, inv_scale, s_grid);
      __builtin_nontemporal_store(r, xout + vec0 + i * BLOCK_THREADS);
    }
  }
#else
  // ---- fallback: NT b128 loads + gfx1250 global_prefetch_b8 ----
#pragma unroll
  for (int i = 0; i < VECS_PER_THREAD; ++i) {
    const int vi = vec0 + i * BLOCK_THREADS;
    if (vi < vec_per_row) {
      const v4f* gp = xin + vi;
      if (i + 1 < VECS_PER_THREAD)
        __builtin_prefetch((const void*)(gp + BLOCK_THREADS), 0, 0);
      v4f v = __builtin_nontemporal_load(gp);
      v4f r = snap4(v, mulA, addB, inv_scale, s_grid);
      __builtin_nontemporal_store(r, xout + vi);
    }
  }
#endif
}

extern "C" void kernel_launch(void* const* d_in, const int* in_sizes, int n_in,
                              void* d_out, int out_size, void* d_ws, size_t ws_size,
                              hipStream_t stream) {
  const float* x     = (const float*)d_in[0];  // [C*F]
  const float* alpha = (const float*)d_in[1];  // [C]
  const float* grid  = (const float*)d_in[2];  // [16]
  float* out = (float*)d_out;

  const int C = in_sizes[1];                   // 4096
  const long long total = (long long)in_sizes[0];
  const int F = (int)(total / (long long)C);   // 16384
  const int vec_per_row = F / 4;               // 4096 float4 per row
  int chunks = (vec_per_row + VEC_PER_BLOCK - 1) / VEC_PER_BLOCK;  // 4
  if (chunks < 1) chunks = 1;

  dim3 g((unsigned)chunks, (unsigned)C, 1);
  quant_dequant_kernel<<<g, dim3(BLOCK_THREADS, 1, 1), 0, stream>>>(
      x, alpha, grid, out, vec_per_row);
}